// Coref_Injection_54185307406634
// MI455X (gfx1250) — compile-verified
//
#include <hip/hip_runtime.h>
#include <hip/hip_bf16.h>

#define Bb   16
#define Pp   1024
#define Ll   512
#define Mm   128
#define Dd   256
#define DISZ 20
#define KKK  788      // 3*D + DIS
#define KPAD 800      // padded feat row (zeros in 788..799)
#define NEGV (-1e30f)
#define BP   (Bb * Pp)   // 16384

typedef __attribute__((ext_vector_type(2))) float v2f;
typedef __attribute__((ext_vector_type(8))) float v8f;

// ---------------- K1: recover one-hot indices, build feat rows + tail_rep ----
__global__ __launch_bounds__(256)
void k_feat(const float* __restrict__ head, const float* __restrict__ tail,
            const int* __restrict__ diss, const float* __restrict__ input,
            const float* __restrict__ dis_embed,
            float* __restrict__ feat, float* __restrict__ tail_rep)
{
    int r = blockIdx.x;            // b*P + p
    int b = r / Pp;
    int t = threadIdx.x;           // 0..255 == D
    __shared__ int s_h, s_t;
    const float* hrow = head + (size_t)r * Ll;
    const float* trow = tail + (size_t)r * Ll;
    for (int l = t; l < Ll; l += 256) {
        if (hrow[l] != 0.0f) s_h = l;   // exactly one nonzero (one-hot)
        if (trow[l] != 0.0f) s_t = l;
    }
    __syncthreads();
    const float* hin = input + ((size_t)b * Ll + s_h) * Dd;
    const float* tin = input + ((size_t)b * Ll + s_t) * Dd;
    float hv = hin[t], tv = tin[t];
    float* frow = feat + (size_t)r * KPAD;
    frow[t]          = hv;
    frow[Dd + t]     = tv;
    frow[2 * Dd + t] = hv * tv;
    tail_rep[(size_t)r * Dd + t] = tv;
    if (t < DISZ)                       frow[3 * Dd + t] = dis_embed[diss[r] * DISZ + t];
    else if (3 * Dd + t < KPAD)         frow[3 * Dd + t] = 0.0f;  // pad 788..799
}

// ---------------- K2: hidden = relu(feat @ W1 + b1) via fp32 WMMA ------------
// wave computes a 16(M) x 64(N) tile; 1024 Mtiles x 4 Ngroups = 4096 waves.
__global__ __launch_bounds__(256)
void k_gemm(const float* __restrict__ feat, const float* __restrict__ W1,
            const float* __restrict__ b1, float* __restrict__ hidden)
{
    int wid  = (blockIdx.x * 256 + threadIdx.x) >> 5;
    int lane = threadIdx.x & 31;
    int mt = wid >> 2;            // 0..1023
    int ng = wid & 3;             // 0..3
    int row0 = mt * 16;
    int n0   = ng * 64;
    int half = lane >> 4;         // 0: K=k0..k0+1 ; 1: K=k0+2..k0+3
    int l16  = lane & 15;

    v8f acc[4] = {v8f{}, v8f{}, v8f{}, v8f{}};
    const float* arow = feat + (size_t)(row0 + l16) * KPAD + half * 2;

    for (int k0 = 0; k0 < KKK; k0 += 4) {
        v2f a;
        a.x = arow[k0];
        a.y = arow[k0 + 1];
        const float* bp = W1 + (size_t)(k0 + half * 2) * Dd + n0 + l16;
#pragma unroll
        for (int nt = 0; nt < 4; ++nt) {
            v2f bv;
            bv.x = bp[nt * 16];
            bv.y = bp[nt * 16 + Dd];
            acc[nt] = __builtin_amdgcn_wmma_f32_16x16x4_f32(
                false, a, false, bv, (short)0, acc[nt], false, false);
        }
    }

#pragma unroll
    for (int nt = 0; nt < 4; ++nt) {
        int col = n0 + nt * 16 + l16;
        float bias = b1[col];
#pragma unroll
        for (int vg = 0; vg < 8; ++vg) {
            int row = row0 + half * 8 + vg;
            float v = acc[nt][vg] + bias;
            hidden[(size_t)row * Dd + col] = v > 0.0f ? v : 0.0f;
        }
    }
}

// ---------------- K3: logits = hidden@W2+b2, KL partials ---------------------
__global__ __launch_bounds__(256)
void k_logits(const float* __restrict__ hidden, const float* __restrict__ W2,
              const float* __restrict__ b2, const float* __restrict__ label,
              const int* __restrict__ mask,
              float* __restrict__ logits1, float* __restrict__ klsum,
              float* __restrict__ msum)
{
    int tid = threadIdx.x;
    int r = blockIdx.x * 256 + tid;
    const float* hrow = hidden + (size_t)r * Dd;
    float l0 = b2[0], l1 = b2[1];
    for (int d = 0; d < Dd; ++d) {
        float h = hrow[d];
        l0 = fmaf(h, W2[d * 2 + 0], l0);
        l1 = fmaf(h, W2[d * 2 + 1], l1);
    }
    logits1[r] = l1;
    float mx  = fmaxf(l0, l1);
    float lse = mx + __logf(__expf(l0 - mx) + __expf(l1 - mx));
    float lq0 = l0 - lse, lq1 = l1 - lse;
    float y0 = label[r * 2 + 0], y1 = label[r * 2 + 1];
    float kl = (y0 > 0.0f ? y0 * __logf(y0) : 0.0f) - y0 * lq0
             + (y1 > 0.0f ? y1 * __logf(y1) : 0.0f) - y1 * lq1;
    float m = mask[r] ? 1.0f : 0.0f;

    __shared__ float sk[256], sm[256];
    sk[tid] = kl * m;
    sm[tid] = m;
    __syncthreads();
    for (int s = 128; s > 0; s >>= 1) {
        if (tid < s) { sk[tid] += sk[tid + s]; sm[tid] += sm[tid + s]; }
        __syncthreads();
    }
    if (tid == 0) { klsum[blockIdx.x] = sk[0]; msum[blockIdx.x] = sm[0]; }
}

// ---------------- K4: per-mention segment max/argmax + coref_rep -------------
__global__ __launch_bounds__(256)
void k_seg(const int* __restrict__ lens, const float* __restrict__ logits1,
           const float* __restrict__ tail_rep, float* __restrict__ coref_rep)
{
    int bm = blockIdx.x;          // b*M + m
    int b = bm / Mm, m = bm % Mm;
    __shared__ int s_sel, s_any;
    __shared__ float s_maxp;
    if (threadIdx.x == 0) {
        int start = 0;
        for (int j = 0; j < m; ++j) start += lens[b * Mm + j];
        int len = lens[b * Mm + m];
        float best = NEGV; int bi = 0; int any = 0;
        const float* lp = logits1 + (size_t)b * Pp;
        for (int i = 0; i < len; ++i) {
            int p = start + i;
            if (p >= 0 && p < Pp) {
                any = 1;
                float v = lp[p];
                if (v > best) { best = v; bi = p; }   // first-occurrence argmax
            }
        }
        s_sel = bi; s_maxp = best; s_any = any;
    }
    __syncthreads();
    int d = threadIdx.x;
    float v = s_any ? tail_rep[((size_t)b * Pp + s_sel) * Dd + d] * s_maxp : 0.0f;
    coref_rep[((size_t)b * Mm + m) * Dd + d] = v;
}

// ---------------- K5: output = input + one-hot-scatter(coref_rep) ------------
__global__ __launch_bounds__(256)
void k_out(const float* __restrict__ input, const float* __restrict__ pos,
           const float* __restrict__ coref_rep, float* __restrict__ out)
{
    int bl = blockIdx.x;          // b*L + l
    int b = bl / Ll, l = bl % Ll;
    __shared__ float spm[Mm];
    if (threadIdx.x < Mm)
        spm[threadIdx.x] = pos[((size_t)b * Mm + threadIdx.x) * Ll + l];
    __syncthreads();
    int d = threadIdx.x;
    float s = input[(size_t)bl * Dd + d];
    for (int m = 0; m < Mm; ++m) {
        float pm = spm[m];                 // uniform across the wave
        if (pm != 0.0f) s = fmaf(pm, coref_rep[((size_t)b * Mm + m) * Dd + d], s);
    }
    out[(size_t)bl * Dd + d] = s;
}

// ---------------- K6: final loss reduction -----------------------------------
__global__ __launch_bounds__(64)
void k_loss(const float* __restrict__ klsum, const float* __restrict__ msum,
            float* __restrict__ out_loss)
{
    __shared__ float sk[64], sm[64];
    int t = threadIdx.x;
    sk[t] = klsum[t];
    sm[t] = msum[t];
    __syncthreads();
    for (int s = 32; s > 0; s >>= 1) {
        if (t < s) { sk[t] += sk[t + s]; sm[t] += sm[t + s]; }
        __syncthreads();
    }
    if (t == 0) out_loss[0] = sk[0] / fmaxf(sm[0], 1.0f);
}

// ---------------- launch -----------------------------------------------------
extern "C" void kernel_launch(void* const* d_in, const int* in_sizes, int n_in,
                              void* d_out, int out_size, void* d_ws, size_t ws_size,
                              hipStream_t stream)
{
    const float* head  = (const float*)d_in[0];
    const float* tail  = (const float*)d_in[1];
    const int*   diss  = (const int*)  d_in[2];
    const int*   lens  = (const int*)  d_in[3];
    const float* input = (const float*)d_in[4];
    const float* cpos  = (const float*)d_in[5];
    const float* label = (const float*)d_in[6];
    const int*   mask  = (const int*)  d_in[7];
    const float* demb  = (const float*)d_in[8];
    const float* W1    = (const float*)d_in[9];
    const float* b1    = (const float*)d_in[10];
    const float* W2    = (const float*)d_in[11];
    const float* b2    = (const float*)d_in[12];

    float* ws = (float*)d_ws;
    float* feat     = ws;                                   // BP * KPAD
    float* tail_rep = feat     + (size_t)BP * KPAD;         // BP * D
    float* hidden   = tail_rep + (size_t)BP * Dd;           // BP * D
    float* logits1  = hidden   + (size_t)BP * Dd;           // BP
    float* klsum    = logits1  + BP;                        // 64
    float* msum     = klsum    + 64;                        // 64
    float* coref    = msum     + 64;                        // B*M*D

    float* out    = (float*)d_out;                          // [B,L,D]
    float* out_ls = out + (size_t)Bb * Ll * Dd;             // scalar loss

    k_feat  <<<BP, 256, 0, stream>>>(head, tail, diss, input, demb, feat, tail_rep);
    k_gemm  <<<(BP / 16) * 4 / 8, 256, 0, stream>>>(feat, W1, b1, hidden);
    k_logits<<<BP / 256, 256, 0, stream>>>(hidden, W2, b2, label, mask,
                                           logits1, klsum, msum);
    k_seg   <<<Bb * Mm, 256, 0, stream>>>(lens, logits1, tail_rep, coref);
    k_out   <<<Bb * Ll, 256, 0, stream>>>(input, cpos, coref, out);
    k_loss  <<<1, 64, 0, stream>>>(klsum, msum, out_ls);
}